// SimpleRotor_1082331758824
// MI455X (gfx1250) — compile-verified
//
#include <hip/hip_runtime.h>
#include <hip/hip_bf16.h>
#include <math.h>

typedef __attribute__((ext_vector_type(2))) float v2f;
typedef __attribute__((ext_vector_type(8))) float v8f;

#define DIM   2048
#define RANK  4
#define NCLS  3

// ---------------------------------------------------------------------------
// Setup: collapse expm of the rank-8 skew generator to an 8x8 problem and
// bake everything into WpadT[16][DIM] (Wfull^T zero-padded to 16 columns).
// ---------------------------------------------------------------------------
__global__ __launch_bounds__(256)
void rotor_setup_kernel(const float* __restrict__ U, const float* __restrict__ V,
                        const float* __restrict__ W, float* __restrict__ WpadT) {
  __shared__ float sG[64];        // G = F^T F   (8x8)
  __shared__ float sH[NCLS * 8];  // H = W  F    (3x8)
  __shared__ float sN[8 * NCLS];  // N           (8x3)
  const int t = threadIdx.x;
  if (t < 64) sG[t] = 0.f;
  if (t < NCLS * 8) sH[t] = 0.f;
  __syncthreads();

  // Partial Gram / projection sums over the 2048 rows of F = [U V].
  float g[64], h[NCLS * 8];
  for (int i = 0; i < 64; ++i) g[i] = 0.f;
  for (int i = 0; i < NCLS * 8; ++i) h[i] = 0.f;
  for (int i = t; i < DIM; i += 256) {
    float f[8];
    for (int c = 0; c < RANK; ++c) { f[c] = U[i * RANK + c]; f[c + 4] = V[i * RANK + c]; }
    for (int a = 0; a < 8; ++a)
      for (int b2 = 0; b2 < 8; ++b2)
        g[a * 8 + b2] += f[a] * f[b2];
    for (int w = 0; w < NCLS; ++w) {
      const float wv = W[w * DIM + i];
      for (int c = 0; c < 8; ++c) h[w * 8 + c] += wv * f[c];
    }
  }
  for (int i = 0; i < 64; ++i) atomicAdd(&sG[i], g[i]);
  for (int i = 0; i < NCLS * 8; ++i) atomicAdd(&sH[i], h[i]);
  __syncthreads();

  if (t == 0) {
    // M = K*G with K = [[0,I],[-I,0]]
    float M[64], T[64], Tn[64], Phi[64];
    for (int r = 0; r < 8; ++r)
      for (int c = 0; c < 8; ++c)
        M[r * 8 + c] = (r < 4) ? sG[(r + 4) * 8 + c] : -sG[(r - 4) * 8 + c];
    // Phi = sum_{p>=0} M^p / (p+1)!   (||M|| ~ 0.2 -> fast convergence)
    for (int i = 0; i < 64; ++i) { T[i] = ((i % 9) == 0) ? 1.f : 0.f; Phi[i] = T[i]; }
    float coef = 1.f;
    for (int p = 1; p <= 24; ++p) {
      for (int r = 0; r < 8; ++r)
        for (int c = 0; c < 8; ++c) {
          float s = 0.f;
          for (int k = 0; k < 8; ++k) s += T[r * 8 + k] * M[k * 8 + c];
          Tn[r * 8 + c] = s;
        }
      for (int i = 0; i < 64; ++i) T[i] = Tn[i];
      coef /= (float)(p + 1);
      for (int i = 0; i < 64; ++i) Phi[i] += coef * T[i];
    }
    // A1 = Phi^T * H^T (8x3);  N = -K*A1
    float A1[8 * NCLS];
    for (int r = 0; r < 8; ++r)
      for (int c = 0; c < NCLS; ++c) {
        float s = 0.f;
        for (int k = 0; k < 8; ++k) s += Phi[k * 8 + r] * sH[c * 8 + k];
        A1[r * NCLS + c] = s;
      }
    for (int r = 0; r < 8; ++r)
      for (int c = 0; c < NCLS; ++c)
        sN[r * NCLS + c] = (r < 4) ? -A1[(r + 4) * NCLS + c] : A1[(r - 4) * NCLS + c];
  }
  __syncthreads();

  // WpadT[n][k]: columns 0..2 hold Wfull = W^T + F*N, columns 3..15 are zero.
  for (int k = t; k < DIM; k += 256) {
    float f[8];
    for (int c = 0; c < RANK; ++c) { f[c] = U[k * RANK + c]; f[c + 4] = V[k * RANK + c]; }
    for (int n = 0; n < NCLS; ++n) {
      float s = W[n * DIM + k];
      for (int c = 0; c < 8; ++c) s += f[c] * sN[c * NCLS + n];
      WpadT[n * DIM + k] = s;
    }
    for (int n = NCLS; n < 16; ++n) WpadT[n * DIM + k] = 0.f;
  }
}

// ---------------------------------------------------------------------------
// Main: one wave32 per 16 rows of x. 512 x V_WMMA_F32_16X16X4_F32 along K,
// then bias + 3-way softmax via an LDS-staged 16x16 D tile.
// ---------------------------------------------------------------------------
__global__ __launch_bounds__(32)
void rotor_main_kernel(const float* __restrict__ x, const float* __restrict__ WpadT,
                       const float* __restrict__ bvec, float* __restrict__ out) {
  const int lane = threadIdx.x;        // 0..31
  const int r0   = blockIdx.x * 16;
  const int n    = lane & 15;          // A: row M; B: column N
  const int kh   = (lane >> 4) * 2;    // half-wave K offset {0,2}

  const float* xrow = x + (size_t)(r0 + n) * DIM + kh;
  const float* wcol = WpadT + (size_t)n * DIM + kh;

  v8f acc = {};
#pragma unroll 4
  for (int k = 0; k < DIM; k += 4) {
    v2f a = *(const v2f*)(xrow + k);                 // x[M][k+kh .. k+kh+1]
    v2f b = *(const v2f*)(wcol + k);                 // Wfull[k+kh..][N]
    __builtin_prefetch(xrow + k + 512, 0, 0);        // stream x ahead (global_prefetch_b8)
    // D = A(16x4) * B(4x16) + C   -> v_wmma_f32_16x16x4_f32
    acc = __builtin_amdgcn_wmma_f32_16x16x4_f32(
        /*neg_a=*/false, a, /*neg_b=*/false, b,
        /*c_mod=*/(short)0, acc, /*reuse_a=*/false, /*reuse_b=*/false);
  }

  // D layout: VGPR j holds (M = j + 8*(lane>>4), N = lane&15)
  __shared__ float tile[16 * 16];
  const int hhi = lane >> 4;
#pragma unroll
  for (int j = 0; j < 8; ++j)
    tile[(j + 8 * hhi) * 16 + n] = acc[j];
  __syncthreads();

  if (lane < 16) {
    const int row = r0 + lane;
    float l0 = tile[lane * 16 + 0] + bvec[0];
    float l1 = tile[lane * 16 + 1] + bvec[1];
    float l2 = tile[lane * 16 + 2] + bvec[2];
    float m  = fmaxf(l0, fmaxf(l1, l2));
    float e0 = __expf(l0 - m), e1 = __expf(l1 - m), e2 = __expf(l2 - m);
    float inv = 1.f / (e0 + e1 + e2);
    out[row * 3 + 0] = e0 * inv;
    out[row * 3 + 1] = e1 * inv;
    out[row * 3 + 2] = e2 * inv;
  }
}

extern "C" void kernel_launch(void* const* d_in, const int* in_sizes, int n_in,
                              void* d_out, int out_size, void* d_ws, size_t ws_size,
                              hipStream_t stream) {
  const float* x = (const float*)d_in[0];
  const float* U = (const float*)d_in[1];
  const float* V = (const float*)d_in[2];
  const float* W = (const float*)d_in[3];
  const float* b = (const float*)d_in[4];
  float* WpadT   = (float*)d_ws;            // 16 * 2048 floats = 128 KB scratch
  float* out     = (float*)d_out;

  rotor_setup_kernel<<<1, 256, 0, stream>>>(U, V, W, WpadT);
  rotor_main_kernel<<<16384 / 16, 32, 0, stream>>>(x, WpadT, b, out);
}